// DiffeqSolver_Attention_26809185861946
// MI455X (gfx1250) — compile-verified
//
#include <hip/hip_runtime.h>
#include <hip/hip_bf16.h>

// ---------------------------------------------------------------------------
// DiffeqSolver_Attention for MI455X (gfx1250, wave32, WMMA f32 16x16x4)
//   B=32, N_TRAJ=512, LATENTS=256, HIDDEN=1024, T=10
//   ODE func (36 evals): Y = tanh(X @ W1 + b1) @ W2, X = (24576, 256)
//   W panels double-buffered into LDS via global_load_async_to_lds_b128.
// ---------------------------------------------------------------------------

typedef __attribute__((ext_vector_type(2))) float v2f;
typedef __attribute__((ext_vector_type(8))) float v8f;

#if defined(__HIP_DEVICE_COMPILE__)
#if !__has_builtin(__builtin_amdgcn_wmma_f32_16x16x4_f32)
#error "missing __builtin_amdgcn_wmma_f32_16x16x4_f32 on this target"
#endif
#endif

#define NB     32
#define NTRAJ  512
#define DLAT   256
#define NHID   1024
#define NTIME  10
#define NX     768                 // NTRAJ + DLAT
#define MTOT   (NB * NX)           // 24576
#define SELEMS ((long)MTOT * DLAT) // 6,291,456 floats per state buffer

#define M_TILE 32
#define XS_PITCH (DLAT + 4)        // 260: conflict-free strided A-frag reads
#define HS_PITCH (NHID + 4)        // 1028
#define PANEL_FLOATS 16384         // one 64KB W panel (16x1024 or 64x256)
#define PANEL_BYTES  65536

__device__ __forceinline__ float fast_tanh(float x) {
    x = fminf(fmaxf(x, -10.0f), 10.0f);
    float e = __expf(2.0f * x);
    return (e - 1.0f) / (e + 1.0f);
}

__device__ __forceinline__ v8f wmma_f32(v2f a, v2f b, v8f c) {
    return __builtin_amdgcn_wmma_f32_16x16x4_f32(false, a, false, b,
                                                 (short)0, c, false, false);
}

// low 32 bits of a generic pointer to LDS == LDS byte address
__device__ __forceinline__ unsigned lds_addr(const void* p) {
    return (unsigned)(unsigned long long)p;
}

// ASYNCcnt-tracked bulk copy: 16 contiguous bytes global -> LDS (GVS mode)
__device__ __forceinline__ void async_b128(unsigned lds_byte, const void* sbase,
                                           int voff_byte) {
    asm volatile("global_load_async_to_lds_b128 %0, %1, %2"
                 :: "v"(lds_byte), "v"(voff_byte), "s"(sbase) : "memory");
}

__device__ __forceinline__ void wait_async_16() {
    asm volatile("s_wait_asynccnt 0x10" ::: "memory");
}
__device__ __forceinline__ void wait_async_0() {
    asm volatile("s_wait_asynccnt 0x0" ::: "memory");
}

// 64KB contiguous panel of W starting at byte kbase -> LDS panel buffer
__device__ __forceinline__ void load_panel_async(unsigned lds_base, const float* W,
                                                 int kbase, int tid) {
#pragma unroll
    for (int i = 0; i < 16; ++i) {
        const int c = (tid + i * 256) * 16;     // byte chunk within panel
        async_b128(lds_base + c, W, kbase + c);
    }
}

// ---------------------------------------------------------------------------
// Fused ODE function: Y = tanh(X @ W1 + b1) @ W2
//   grid.x = 768 blocks, 256 threads (8 waves), dynamic LDS = 295,936 B
// ---------------------------------------------------------------------------
__global__ __launch_bounds__(256, 1)
void ode_func_kernel(const float* __restrict__ Xin,
                     const float* __restrict__ W1,
                     const float* __restrict__ b1,
                     const float* __restrict__ W2,
                     float* __restrict__ Y) {
    extern __shared__ float smem[];
    float* Xs = smem;                                   // [32][260]
    float* Hs = Xs + M_TILE * XS_PITCH;                 // [32][1028]
    float* Wp = Hs + M_TILE * HS_PITCH;                 // 2 x 16384 (panels)

    const int tid  = threadIdx.x;
    const int lane = tid & 31;
    const int wave = tid >> 5;
    const int l16  = lane & 15;
    const int half = lane >> 4;
    const long m0  = (long)blockIdx.x * M_TILE;

    const unsigned XsA = lds_addr(Xs);
    const unsigned WpA = lds_addr(Wp);

    // ---- prologue: async-stage X tile (32x256, padded pitch) + W1 panel 0 --
#pragma unroll
    for (int i = 0; i < 8; ++i) {
        const int c  = tid + i * 256;       // float4 chunk id, 0..2047
        const int r  = c >> 6;              // row 0..31
        const int c4 = c & 63;              // float4 col
        async_b128(XsA + (unsigned)(r * XS_PITCH + c4 * 4) * 4,
                   Xin, (int)((m0 + r) * DLAT + c4 * 4) * 4);
    }
    load_panel_async(WpA, W1, 0, tid);

    // ---- phase 1: Hs = tanh(X @ W1 + b1) ------------------------------------
    // K = 256 as 16 k-blocks of 16; wave owns 8 n-tiles x 2 m-subtiles.
    {
        v8f acc0[8] = {};
        v8f acc1[8] = {};
        for (int kb = 0; kb < 16; ++kb) {
            if (kb + 1 < 16) {
                load_panel_async(WpA + ((kb + 1) & 1) * PANEL_BYTES, W1,
                                 (kb + 1) * PANEL_BYTES, tid);
                wait_async_16();            // in-order: panel kb (and Xs) done
            } else {
                wait_async_0();
            }
            __syncthreads();                // panel kb visible to all waves
            const float* P = Wp + (kb & 1) * PANEL_FLOATS;
#pragma unroll
            for (int kk = 0; kk < 4; ++kk) {
                const int kl = kk * 4 + 2 * half;     // panel-local k row
                const int kg = kb * 16 + kl;          // Xs column
                const float* xr0 = Xs + l16 * XS_PITCH + kg;
                const float* xr1 = Xs + (16 + l16) * XS_PITCH + kg;
                v2f a0; a0.x = xr0[0]; a0.y = xr0[1];
                v2f a1; a1.x = xr1[0]; a1.y = xr1[1];
#pragma unroll
                for (int j = 0; j < 8; ++j) {
                    const int n0 = (wave * 8 + j) * 16;
                    v2f b;
                    b.x = P[kl * NHID + n0 + l16];
                    b.y = P[(kl + 1) * NHID + n0 + l16];
                    acc0[j] = wmma_f32(a0, b, acc0[j]);
                    acc1[j] = wmma_f32(a1, b, acc1[j]);
                }
            }
            __syncthreads();                // all waves done reading panel kb
        }
#pragma unroll
        for (int j = 0; j < 8; ++j) {
            const int n0 = (wave * 8 + j) * 16;
            const float bias = b1[n0 + l16];
#pragma unroll
            for (int r = 0; r < 8; ++r) {
                const int row = r + 8 * half;
                Hs[row * HS_PITCH + n0 + l16]        = fast_tanh(acc0[j][r] + bias);
                Hs[(16 + row) * HS_PITCH + n0 + l16] = fast_tanh(acc1[j][r] + bias);
            }
        }
    }
    __syncthreads();

    // ---- phase 2: Y = Hs @ W2 ----------------------------------------------
    // K = 1024 as 16 k-blocks of 64; wave owns 2 n-tiles x 2 m-subtiles.
    {
        load_panel_async(WpA, W2, 0, tid);
        v8f c00 = {}, c01 = {}, c10 = {}, c11 = {};
        const int na  = wave * 32 + l16;        // n-tile 0 column
        const int nb2 = wave * 32 + 16 + l16;   // n-tile 1 column
        for (int kb = 0; kb < 16; ++kb) {
            if (kb + 1 < 16) {
                load_panel_async(WpA + ((kb + 1) & 1) * PANEL_BYTES, W2,
                                 (kb + 1) * PANEL_BYTES, tid);
                wait_async_16();
            } else {
                wait_async_0();
            }
            __syncthreads();
            const float* P = Wp + (kb & 1) * PANEL_FLOATS;
#pragma unroll
            for (int kk = 0; kk < 16; ++kk) {
                const int kl = kk * 4 + 2 * half;     // 0..63 within panel
                const int kg = kb * 64 + kl;          // Hs column
                const float* hr0 = Hs + l16 * HS_PITCH + kg;
                const float* hr1 = Hs + (16 + l16) * HS_PITCH + kg;
                v2f a0; a0.x = hr0[0]; a0.y = hr0[1];
                v2f a1; a1.x = hr1[0]; a1.y = hr1[1];
                v2f b0, b1v;
                b0.x  = P[kl * DLAT + na];
                b0.y  = P[(kl + 1) * DLAT + na];
                b1v.x = P[kl * DLAT + nb2];
                b1v.y = P[(kl + 1) * DLAT + nb2];
                c00 = wmma_f32(a0, b0, c00);
                c10 = wmma_f32(a1, b0, c10);
                c01 = wmma_f32(a0, b1v, c01);
                c11 = wmma_f32(a1, b1v, c11);
            }
            __syncthreads();
        }
#pragma unroll
        for (int r = 0; r < 8; ++r) {
            const int row = r + 8 * half;
            const int n0 = wave * 32;
            Y[(m0 + row) * DLAT + n0 + l16]           = c00[r];
            Y[(m0 + 16 + row) * DLAT + n0 + l16]      = c10[r];
            Y[(m0 + row) * DLAT + n0 + 16 + l16]      = c01[r];
            Y[(m0 + 16 + row) * DLAT + n0 + 16 + l16] = c11[r];
        }
    }
}

// ---------------------------------------------------------------------------
// mean over trajectories: mean[b][d] = (1/512) sum_n X[b][n][d]
// ---------------------------------------------------------------------------
__global__ __launch_bounds__(256)
void mean_kernel(const float* __restrict__ X, float* __restrict__ mean) {
    const int idx = blockIdx.x * 256 + threadIdx.x;   // 0..8191
    const int b = idx >> 8, d = idx & 255;
    const float* p = X + ((long)b * NTRAJ) * DLAT + d;
    float s = 0.0f;
    for (int n = 0; n < NTRAJ; ++n) s += p[(long)n * DLAT];
    mean[idx] = s * (1.0f / (float)NTRAJ);
}

// ---------------------------------------------------------------------------
// cov[b] = Xm^T Xm  (256x512 * 512x256), WMMA f32, K=512
// ---------------------------------------------------------------------------
__global__ __launch_bounds__(256, 1)
void cov_kernel(const float* __restrict__ X, const float* __restrict__ mean,
                float* __restrict__ cov) {
    const int b    = blockIdx.y;
    const int m0   = blockIdx.x * 32;
    const int tid  = threadIdx.x;
    const int lane = tid & 31;
    const int wave = tid >> 5;
    const int l16  = lane & 15;
    const int half = lane >> 4;

    const float* Xb = X + (long)b * NTRAJ * DLAT;
    const float* mb = mean + b * DLAT;
    const float am0 = mb[m0 + l16];
    const float am1 = mb[m0 + 16 + l16];

    for (int j = 0; j < 2; ++j) {
        const int n0 = (wave * 2 + j) * 16;
        const float bm = mb[n0 + l16];
        v8f acc0 = {};
        v8f acc1 = {};
        for (int k0 = 0; k0 < NTRAJ; k0 += 4) {
            const int ka = k0 + 2 * half;
            const float* r0 = Xb + (long)(ka + 0) * DLAT;
            const float* r1 = Xb + (long)(ka + 1) * DLAT;
            v2f a0, a1, bb;
            a0.x = r0[m0 + l16] - am0;      a0.y = r1[m0 + l16] - am0;
            a1.x = r0[m0 + 16 + l16] - am1; a1.y = r1[m0 + 16 + l16] - am1;
            bb.x = r0[n0 + l16] - bm;       bb.y = r1[n0 + l16] - bm;
            acc0 = wmma_f32(a0, bb, acc0);
            acc1 = wmma_f32(a1, bb, acc1);
        }
        float* cb = cov + (long)b * DLAT * DLAT;
#pragma unroll
        for (int r = 0; r < 8; ++r) {
            const int row = r + 8 * half;
            cb[(m0 + row) * DLAT + n0 + l16]      = acc0[r];
            cb[(m0 + 16 + row) * DLAT + n0 + l16] = acc1[r];
        }
    }
}

// rstd[b][i] = rsqrt(cov[b][i][i])
__global__ __launch_bounds__(256)
void rstd_kernel(const float* __restrict__ cov, float* __restrict__ rstd) {
    const int idx = blockIdx.x * 256 + threadIdx.x;
    const int b = idx >> 8, i = idx & 255;
    rstd[idx] = rsqrtf(cov[((long)b * DLAT + i) * DLAT + i]);
}

// xx[b, 0:512, :] = first_point (also emitted to out at t=0)
// xx[b, 512:768, :] = corr
__global__ __launch_bounds__(256)
void build_xx_kernel(const float* __restrict__ fp, const float* __restrict__ cov,
                     const float* __restrict__ rstd, float* __restrict__ x,
                     float* __restrict__ out) {
    const long i = (long)blockIdx.x * 256 + threadIdx.x;   // over MTOT*DLAT
    const int d = (int)(i & 255);
    const long rb = i >> 8;
    const int row = (int)(rb % NX);
    const int b   = (int)(rb / NX);
    float v;
    if (row < NTRAJ) {
        v = fp[((long)b * NTRAJ + row) * DLAT + d];
        out[(((long)b * NTRAJ + row) * NTIME + 0) * DLAT + d] = v;
    } else {
        const int r = row - NTRAJ;
        v = cov[((long)b * DLAT + r) * DLAT + d] * rstd[b * DLAT + r] * rstd[b * DLAT + d];
    }
    x[i] = v;
}

// xtmp = x + (ct*dt)*k ; xacc = accIn + (ca*dt)*k   (float4, one pass)
__global__ __launch_bounds__(256)
void rk_combine_kernel(float* __restrict__ xtmp, float* __restrict__ xacc,
                       const float* __restrict__ x, const float* __restrict__ accIn,
                       const float* __restrict__ k, const float* __restrict__ ts,
                       int t0, int t1, float ct, float ca) {
    const float dt = ts[t1] - ts[t0];
    const float bt = ct * dt, ba = ca * dt;
    const long i = ((long)blockIdx.x * 256 + threadIdx.x) * 4;
    const float4 xv = *(const float4*)(x + i);
    const float4 av = *(const float4*)(accIn + i);
    const float4 kv = *(const float4*)(k + i);
    float4 tv, cv;
    tv.x = fmaf(bt, kv.x, xv.x); tv.y = fmaf(bt, kv.y, xv.y);
    tv.z = fmaf(bt, kv.z, xv.z); tv.w = fmaf(bt, kv.w, xv.w);
    cv.x = fmaf(ba, kv.x, av.x); cv.y = fmaf(ba, kv.y, av.y);
    cv.z = fmaf(ba, kv.z, av.z); cv.w = fmaf(ba, kv.w, av.w);
    *(float4*)(xtmp + i) = tv;
    *(float4*)(xacc + i) = cv;
}

// x = xacc + (dt/6)*k ; emit rows < 512 into out[b, n, t, :]   (float4)
__global__ __launch_bounds__(256)
void rk_final_kernel(float* __restrict__ x, const float* __restrict__ xacc,
                     const float* __restrict__ k, const float* __restrict__ ts,
                     int t0, int t1, float* __restrict__ out) {
    const float beta = (1.0f / 6.0f) * (ts[t1] - ts[t0]);
    const long c = (long)blockIdx.x * 256 + threadIdx.x;   // float4 chunk id
    const long i = c * 4;
    const float4 av = *(const float4*)(xacc + i);
    const float4 kv = *(const float4*)(k + i);
    float4 r;
    r.x = fmaf(beta, kv.x, av.x);
    r.y = fmaf(beta, kv.y, av.y);
    r.z = fmaf(beta, kv.z, av.z);
    r.w = fmaf(beta, kv.w, av.w);
    *(float4*)(x + i) = r;

    const int d4  = (int)(c & 63);
    const long rg = c >> 6;
    const int row = (int)(rg % NX);
    const int b   = (int)(rg / NX);
    if (row < NTRAJ) {
        float* o = out + ((((long)b * NTRAJ + row) * NTIME + t1) * DLAT) + d4 * 4;
        *(float4*)o = r;
    }
}

// ---------------------------------------------------------------------------
extern "C" void kernel_launch(void* const* d_in, const int* in_sizes, int n_in,
                              void* d_out, int out_size, void* d_ws, size_t ws_size,
                              hipStream_t stream) {
    const float* fp = (const float*)d_in[0];   // (32, 512, 256)
    const float* ts = (const float*)d_in[1];   // (10,)
    const float* W1 = (const float*)d_in[2];   // (256, 1024)
    const float* b1 = (const float*)d_in[3];   // (1024,)
    const float* W2 = (const float*)d_in[4];   // (1024, 256)
    float* out = (float*)d_out;                // (32, 512, 10, 256)

    float* ws = (float*)d_ws;
    float* x    = ws;               // state (MTOT x DLAT)
    float* xtmp = x    + SELEMS;    // RK4 stage input
    float* xacc = xtmp + SELEMS;    // RK4 accumulation
    float* kcur = xacc + SELEMS;    // current k
    float* mean = kcur + SELEMS;    // 32*256
    float* rstd = mean + NB * DLAT; // 32*256
    float* cov  = rstd + NB * DLAT; // 32*256*256

    const int odeGrid = MTOT / M_TILE;                             // 768
    const size_t odeLds =
        (size_t)(M_TILE * XS_PITCH + M_TILE * HS_PITCH + 2 * PANEL_FLOATS)
        * sizeof(float);                                           // 295,936 B
    const int axGrid = (int)(SELEMS / 4 / 256);                    // 6144
    const int xxGrid = (int)(SELEMS / 256);                        // 24576

    // ---- attention (batched corrcoef) + concat into state + emit t=0 ----
    mean_kernel<<<NB * DLAT / 256, 256, 0, stream>>>(fp, mean);
    cov_kernel<<<dim3(8, NB), 256, 0, stream>>>(fp, mean, cov);
    rstd_kernel<<<NB * DLAT / 256, 256, 0, stream>>>(cov, rstd);
    build_xx_kernel<<<xxGrid, 256, 0, stream>>>(fp, cov, rstd, x, out);

    // ---- RK4 time stepping ----
    for (int t = 0; t < NTIME - 1; ++t) {
        // k1: xtmp = x + dt/2 k1 ; xacc = x + dt/6 k1
        ode_func_kernel<<<odeGrid, 256, odeLds, stream>>>(x, W1, b1, W2, kcur);
        rk_combine_kernel<<<axGrid, 256, 0, stream>>>(xtmp, xacc, x, x, kcur,
                                                      ts, t, t + 1, 0.5f, 1.0f / 6.0f);
        // k2: xtmp = x + dt/2 k2 ; xacc += dt/3 k2
        ode_func_kernel<<<odeGrid, 256, odeLds, stream>>>(xtmp, W1, b1, W2, kcur);
        rk_combine_kernel<<<axGrid, 256, 0, stream>>>(xtmp, xacc, x, xacc, kcur,
                                                      ts, t, t + 1, 0.5f, 1.0f / 3.0f);
        // k3: xtmp = x + dt k3 ; xacc += dt/3 k3
        ode_func_kernel<<<odeGrid, 256, odeLds, stream>>>(xtmp, W1, b1, W2, kcur);
        rk_combine_kernel<<<axGrid, 256, 0, stream>>>(xtmp, xacc, x, xacc, kcur,
                                                      ts, t, t + 1, 1.0f, 1.0f / 3.0f);
        // k4: x = xacc + dt/6 k4 ; emit rows<512 at t+1
        ode_func_kernel<<<odeGrid, 256, odeLds, stream>>>(xtmp, W1, b1, W2, kcur);
        rk_final_kernel<<<axGrid, 256, 0, stream>>>(x, xacc, kcur, ts, t, t + 1, out);
    }
}